// IndexedLinearLayer_22239340659214
// MI455X (gfx1250) — compile-verified
//
#include <hip/hip_runtime.h>

// ---------------------------------------------------------------------------
// IndexedLinearLayer for MI455X (gfx1250, wave32, WMMA + async global->LDS).
//
// Effective math (see reference): only indices[-1] matters.
//   W_eff[j,k] = indexed_weights[j, k, indices[BATCH-1, j]]
//   out[b,j]   = sum_k x[b,k] * W_eff[j,k] + bias[j]
//
// Phase 1: gather W_eff (1024x1024 f32) into d_ws (~128 MB HBM, stride-256).
// Phase 2: f32 WMMA GEMM. 128x128 block tile, K-chunks of 32 staged into LDS
//          via GLOBAL_LOAD_ASYNC_TO_LDS_B128 (ASYNCcnt, s_wait_asynccnt),
//          double-buffered so the async engine overlaps the matrix pipe.
//          Per wave: 32x64 tile = 2x4 16x16 tiles, fragments via ds_load_b64.
// ---------------------------------------------------------------------------

typedef __attribute__((ext_vector_type(2))) float v2f;
typedef __attribute__((ext_vector_type(8))) float v8f;

#define SIZE_IN   1024
#define SIZE_OUT  1024
#define NQ        256
#define BATCH     4096
#define KC        32      // K-chunk staged in LDS
#define PAD       36      // padded LDS row stride (floats): 144B = 16B-mult,
                          // and 36*i mod 64 banks are conflict-free for frags

typedef const __attribute__((address_space(1))) void* gas_ptr;   // global (64-bit)
typedef __attribute__((address_space(3)))       void* las_ptr;   // LDS (32-bit)

// Async DMA: copies 16 bytes per lane from global address gsrc to LDS address lds_dst, tracked by ASYNCcnt.
static __device__ __forceinline__ void async_copy_b128(void* lds_dst, const void* gsrc) {
    asm volatile("global_load_async_to_lds_b128 %0, %1, off"
                 :: "v"((las_ptr)lds_dst), "v"((gas_ptr)gsrc)
                 : "memory");
}

static __device__ __forceinline__ void wait_async0() {
    asm volatile("s_wait_asynccnt 0x0" ::: "memory");
}

// -------------------------- Phase 1: gather --------------------------------
// tid = j*SIZE_IN + k  ->  iw[((j*SIZE_IN)+k)*NQ + q_j] = iw[tid*NQ + q_j]
__global__ __launch_bounds__(256) void gather_w_kernel(
    const float* __restrict__ iw,        // [SIZE_OUT, SIZE_IN, NQ]
    const int*   __restrict__ indices,   // [BATCH, SIZE_OUT]
    float*       __restrict__ w)         // [SIZE_OUT, SIZE_IN]  (d_ws)
{
    const int tid = blockIdx.x * 256 + threadIdx.x;     // 0 .. 1M-1
    const int j   = tid >> 10;                          // output row
    const int q   = indices[(size_t)(BATCH - 1) * SIZE_OUT + j];
    w[tid] = iw[(size_t)tid * NQ + (size_t)q];
}

// -------------------------- Phase 2: WMMA GEMM -----------------------------
// Fragment layouts (wave32, 32-bit elements):
//   A 16x4 : lane (hi,l16): VGPR0 = A[l16][k+2*hi], VGPR1 = A[l16][k+2*hi+1]
//   B 4x16 : lane (hi,l16): VGPR0 = B[k+2*hi][l16], VGPR1 = B[k+2*hi+1][l16]
//   C/D 16x16 : VGPR r -> row (r + 8*hi), col l16
// Both A and B fragment reads are 8B ds_load_b64 from the padded LDS tiles.
__global__ __launch_bounds__(256) void wmma_gemm_kernel(
    const float* __restrict__ x,     // [BATCH, SIZE_IN]
    const float* __restrict__ w,     // [SIZE_OUT, SIZE_IN]
    const float* __restrict__ bias,  // [SIZE_OUT]
    float*       __restrict__ out)   // [BATCH, SIZE_OUT]
{
    __shared__ __align__(16) float sx[2][128 * PAD];   // 36 KB
    __shared__ __align__(16) float sw[2][128 * PAD];   // 36 KB

    const int tid  = threadIdx.x;
    const int lane = tid & 31;
    const int wave = tid >> 5;       // 0..7
    const int l16  = lane & 15;
    const int hi   = lane >> 4;      // 0 or 1

    const int row0 = blockIdx.x * 128;   // batch rows of this block
    const int col0 = blockIdx.y * 128;   // output cols of this block

    // Per-thread copy slots: 128 rows x 8 groups of 4 floats = 1024 slots,
    // 4 slots/thread, one b128 async copy each (16B, aligned: PAD*4=144B rows).
    int c_row[4], c_g[4];
    #pragma unroll
    for (int i = 0; i < 4; ++i) {
        const int slot = tid + i * 256;
        c_row[i] = slot >> 3;
        c_g[i]   = slot & 7;
    }

    const float* gx = x + (size_t)row0 * SIZE_IN;   // x tile base
    const float* gw = w + (size_t)col0 * SIZE_IN;   // w tile base

    // Prologue: stage chunk 0 into buffer 0.
    #pragma unroll
    for (int i = 0; i < 4; ++i) {
        async_copy_b128(&sx[0][c_row[i] * PAD + c_g[i] * 4],
                        gx + (size_t)c_row[i] * SIZE_IN + c_g[i] * 4);
        async_copy_b128(&sw[0][c_row[i] * PAD + c_g[i] * 4],
                        gw + (size_t)c_row[i] * SIZE_IN + c_g[i] * 4);
    }

    v8f acc[2][4] = {};   // 32(M) x 64(N) per wave

    // Per-wave fragment bases within the LDS tiles.
    const int aRow = (wave & 3) * 32 + l16;        // A rows (2 frags, +16)
    const int bRow = (wave >> 2) * 64 + l16;       // B rows (4 frags, +16 each)
    const int aOff = aRow * PAD + 2 * hi;
    const int bOff = bRow * PAD + 2 * hi;

    int cur = 0;
    for (int kc = 0; kc < SIZE_IN; kc += KC) {
        wait_async0();        // our async copies for chunk `kc` are done
        __syncthreads();      // => everyone's copies done, prev buffer free

        if (kc + KC < SIZE_IN) {                 // stage next chunk
            const int nb = cur ^ 1;
            const float* gxn = gx + (kc + KC);
            const float* gwn = gw + (kc + KC);
            #pragma unroll
            for (int i = 0; i < 4; ++i) {
                async_copy_b128(&sx[nb][c_row[i] * PAD + c_g[i] * 4],
                                gxn + (size_t)c_row[i] * SIZE_IN + c_g[i] * 4);
                async_copy_b128(&sw[nb][c_row[i] * PAD + c_g[i] * 4],
                                gwn + (size_t)c_row[i] * SIZE_IN + c_g[i] * 4);
            }
        }

        const float* ax = &sx[cur][aOff];
        const float* bw = &sw[cur][bOff];

        #pragma unroll
        for (int kk = 0; kk < KC; kk += 4) {
            v2f a0 = *(const v2f*)(ax + kk);
            v2f a1 = *(const v2f*)(ax + 16 * PAD + kk);
            v2f b0 = *(const v2f*)(bw + kk);
            v2f b1 = *(const v2f*)(bw + 16 * PAD + kk);
            v2f b2 = *(const v2f*)(bw + 32 * PAD + kk);
            v2f b3 = *(const v2f*)(bw + 48 * PAD + kk);

            // (neg_a, A, neg_b, B, c_mod, C, reuse_a, reuse_b)
            acc[0][0] = __builtin_amdgcn_wmma_f32_16x16x4_f32(false, a0, false, b0, (short)0, acc[0][0], false, false);
            acc[0][1] = __builtin_amdgcn_wmma_f32_16x16x4_f32(false, a0, false, b1, (short)0, acc[0][1], false, false);
            acc[0][2] = __builtin_amdgcn_wmma_f32_16x16x4_f32(false, a0, false, b2, (short)0, acc[0][2], false, false);
            acc[0][3] = __builtin_amdgcn_wmma_f32_16x16x4_f32(false, a0, false, b3, (short)0, acc[0][3], false, false);
            acc[1][0] = __builtin_amdgcn_wmma_f32_16x16x4_f32(false, a1, false, b0, (short)0, acc[1][0], false, false);
            acc[1][1] = __builtin_amdgcn_wmma_f32_16x16x4_f32(false, a1, false, b1, (short)0, acc[1][1], false, false);
            acc[1][2] = __builtin_amdgcn_wmma_f32_16x16x4_f32(false, a1, false, b2, (short)0, acc[1][2], false, false);
            acc[1][3] = __builtin_amdgcn_wmma_f32_16x16x4_f32(false, a1, false, b3, (short)0, acc[1][3], false, false);
        }
        cur ^= 1;
    }

    // Epilogue: bias add + store. For tile (m,n): col = col0 + wave_n + 16n + l16 is
    // uniform across the 8 accumulator VGPRs; row = row0 + wave_m + 16m + 8*hi + r.
    const int wRow0 = row0 + (wave & 3) * 32;
    const int wCol0 = col0 + (wave >> 2) * 64;
    #pragma unroll
    for (int m = 0; m < 2; ++m) {
        #pragma unroll
        for (int n = 0; n < 4; ++n) {
            const int   c  = wCol0 + 16 * n + l16;
            const float bv = bias[c];
            float* o = out + (size_t)(wRow0 + 16 * m + 8 * hi) * SIZE_OUT + c;
            #pragma unroll
            for (int r = 0; r < 8; ++r) {
                o[(size_t)r * SIZE_OUT] = acc[m][n][r] + bv;
            }
        }
    }
}

// ---------------------------------------------------------------------------
extern "C" void kernel_launch(void* const* d_in, const int* in_sizes, int n_in,
                              void* d_out, int out_size, void* d_ws, size_t ws_size,
                              hipStream_t stream) {
    (void)in_sizes; (void)n_in; (void)out_size; (void)ws_size;

    const float* x       = (const float*)d_in[0];  // [4096,1024] f32
    const int*   indices = (const int*)  d_in[1];  // [4096,1024] i32
    const float* iw      = (const float*)d_in[2];  // [1024,1024,256] f32
    const float* bias    = (const float*)d_in[3];  // [1024] f32
    float*       out     = (float*)d_out;          // [4096,1024] f32
    float*       w_eff   = (float*)d_ws;           // 1024*1024*4 = 4 MB scratch

    // Phase 1: gather effective weights (fully overwrites the 4 MB region).
    gather_w_kernel<<<(SIZE_OUT * SIZE_IN) / 256, 256, 0, stream>>>(iw, indices, w_eff);

    // Phase 2: WMMA GEMM. Grid: 4096/128 x 1024/128 = 32 x 8 blocks.
    dim3 grid(BATCH / 128, SIZE_OUT / 128);
    wmma_gemm_kernel<<<grid, 256, 0, stream>>>(x, w_eff, bias, out);
}